// BlackBoxV4_14877766713682
// MI455X (gfx1250) — compile-verified
//
#include <hip/hip_runtime.h>
#include <hip/hip_bf16.h>
#include <math.h>

// D=64, K=8, N_INNER=3, VOCAB=32000, B=16, N=256
#define DIM    64
#define KEXP   8
#define NINNER 3
#define VOCAB  32000
#define BATCH  16
#define SEQ    256

typedef __attribute__((ext_vector_type(2))) float v2f;
typedef __attribute__((ext_vector_type(8))) float v8f;

__device__ __forceinline__ v8f wmma_f32_4(v2f a, v2f b, v8f c) {
    // D = A(16x4,f32) * B(4x16,f32) + C(16x16,f32)
    return __builtin_amdgcn_wmma_f32_16x16x4_f32(
        /*neg_a=*/false, a, /*neg_b=*/false, b,
        /*c_mod=*/(short)0, c, /*reuse_a=*/false, /*reuse_b=*/false);
}

__device__ __forceinline__ float gelu_exact(float x) {
    return 0.5f * x * (1.0f + erff(x * 0.70710678118654752440f));
}

// ---------------------------------------------------------------------------
// Kernel 1: recurrent MoE scan. One workgroup, 8 waves, wave w owns expert w.
// States S[k] (16x64 f32) live in LDS; expert weights Ws[k] live in VGPRs as
// WMMA B-tiles. Produces hidden (B, N, D) f32 into workspace.
// ---------------------------------------------------------------------------
__global__ __launch_bounds__(256, 1)
void moe_scan_kernel(const int* __restrict__ ids,
                     const float* __restrict__ embed,
                     const float* __restrict__ Ws,
                     const float* __restrict__ rW,
                     const float* __restrict__ rb,
                     const float* __restrict__ cW,
                     const float* __restrict__ cb,
                     float* __restrict__ hidden)
{
    __shared__ float tok[BATCH][DIM];          // current token        (4 KB)
    __shared__ float S[KEXP][BATCH][DIM];      // expert states        (32 KB)
    __shared__ float wgt[KEXP][BATCH];         // router weights       (0.5 KB)
    __shared__ float comb[BATCH][DIM];         // combiner accumulator (4 KB)

    const int t    = threadIdx.x;
    const int wave = t >> 5;                   // expert id k
    const int lane = t & 31;
    const int nl   = lane & 15;                // N / M index within tile
    const int hi   = lane >> 4;                // half-wave selector

    // ---- zero initial states (covered by barrier at top of step loop) ----
    {
        float4 z = make_float4(0.f, 0.f, 0.f, 0.f);
        float4* Sp = (float4*)&S[0][0][0];     // 2048 float4 / 256 threads
        #pragma unroll
        for (int i = 0; i < 8; ++i) Sp[t + 256 * i] = z;
    }

    // ---- load this expert's weights as WMMA B-tiles (held in VGPRs) ----
    // new[b,e] = sum_d S[b,d] * Wk[e,d]  ->  B[d][e] = Wk[e][d]
    // B-tile (kk,nt): lane -> e = nt*16+nl, k-pair = 4*kk + 2*hi + {0,1}
    v2f wB[4][16];
    {
        const float* Wk = Ws + wave * DIM * DIM;
        #pragma unroll
        for (int nt = 0; nt < 4; ++nt) {
            const int e = nt * 16 + nl;
            #pragma unroll
            for (int kk = 0; kk < 16; ++kk)
                wB[nt][kk] = *(const v2f*)&Wk[e * DIM + 4 * kk + 2 * hi];
        }
    }

    for (int n = 0; n < SEQ; ++n) {
        __syncthreads();   // protects tok/wgt/comb reuse + initial S zeroing

        // ---- gather token: tok[b][:] = embed[ids[b, n]] ----
        {
            const int b  = t >> 4;
            const int c4 = (t & 15) * 4;
            const int id = ids[b * SEQ + n];
            *(float4*)&tok[b][c4] = *(const float4*)&embed[(size_t)id * DIM + c4];
        }
        __syncthreads();   // tok ready

        // ---- router softmax (one thread per batch row) + zero combiner ----
        if (t < BATCH) {
            float lg[KEXP];
            float mx = -3.0e38f;
            #pragma unroll
            for (int k2 = 0; k2 < KEXP; ++k2) {
                float s = rb[k2];
                const float* rwk = rW + k2 * DIM;
                for (int d = 0; d < DIM; ++d) s = fmaf(tok[t][d], rwk[d], s);
                lg[k2] = s;
                mx = fmaxf(mx, s);
            }
            float sum = 0.f;
            #pragma unroll
            for (int k2 = 0; k2 < KEXP; ++k2) { lg[k2] = expf(lg[k2] - mx); sum += lg[k2]; }
            const float inv = 1.0f / sum;
            #pragma unroll
            for (int k2 = 0; k2 < KEXP; ++k2) wgt[k2][t] = lg[k2] * inv;
        }
        ((float4*)comb)[t] = make_float4(0.f, 0.f, 0.f, 0.f);   // 1024 floats
        __syncthreads();   // wgt + comb ready

        // ---- inner recurrence: wave `wave` owns S[wave]; no cross-wave deps
        for (int it = 0; it < NINNER; ++it) {
            v8f acc[4] = {v8f{}, v8f{}, v8f{}, v8f{}};
            #pragma unroll
            for (int kk = 0; kk < 16; ++kk) {
                const v2f a = *(const v2f*)&S[wave][nl][4 * kk + 2 * hi];
                #pragma unroll
                for (int nt = 0; nt < 4; ++nt)
                    acc[nt] = wmma_f32_4(a, wB[nt][kk], acc[nt]);
            }
            // gelu + token add + router blend, write back state
            #pragma unroll
            for (int nt = 0; nt < 4; ++nt) {
                const int e = nt * 16 + nl;
                #pragma unroll
                for (int v = 0; v < 8; ++v) {
                    const int m = v + 8 * hi;
                    const float w   = wgt[wave][m];
                    const float g   = gelu_exact(acc[nt][v] + tok[m][e]);
                    const float old = S[wave][m][e];
                    S[wave][m][e] = w * g + (1.0f - w) * old;
                }
            }
        }

        // ---- combiner: wave k contributes S[k] @ cW[:, k*64:(k+1)*64]^T ----
        {
            v8f cacc[4] = {v8f{}, v8f{}, v8f{}, v8f{}};
            #pragma unroll
            for (int kk = 0; kk < 16; ++kk) {
                const v2f a = *(const v2f*)&S[wave][nl][4 * kk + 2 * hi];
                #pragma unroll
                for (int nt = 0; nt < 4; ++nt) {
                    const int e = nt * 16 + nl;
                    const v2f bb = *(const v2f*)
                        &cW[(size_t)e * (KEXP * DIM) + wave * DIM + 4 * kk + 2 * hi];
                    cacc[nt] = wmma_f32_4(a, bb, cacc[nt]);
                }
            }
            #pragma unroll
            for (int nt = 0; nt < 4; ++nt) {
                const int e = nt * 16 + nl;
                #pragma unroll
                for (int v = 0; v < 8; ++v) {
                    const int m = v + 8 * hi;
                    atomicAdd(&comb[m][e], cacc[nt][v]);   // ds_add_f32
                }
            }
        }
        __syncthreads();   // combiner reduction complete

        // ---- hidden[b, n, :] = comb + combiner_b ----
        {
            const int b  = t >> 4;
            const int c4 = (t & 15) * 4;
            float4 v  = *(const float4*)&comb[b][c4];
            float4 bb = *(const float4*)&cb[c4];
            v.x += bb.x; v.y += bb.y; v.z += bb.z; v.w += bb.w;
            *(float4*)&hidden[((size_t)b * SEQ + n) * DIM + c4] = v;
        }
    }
}

// ---------------------------------------------------------------------------
// Kernel 2: logits = hidden (4096x64) @ output_W^T (64x32000) + output_b.
// Store-bandwidth bound (524 MB out) -> exact f32 WMMA is the right precision.
// Each wave computes a 64(M) x 64(N) block: 4 M-tiles share each B tile, so
// output_W is re-read from L2 only 64x (525 MB, matching the store stream)
// instead of 256x. 64 row-groups x 500 col-groups = 32000 wave-blocks
// = 4000 blocks x 8 waves.
// ---------------------------------------------------------------------------
__global__ __launch_bounds__(256)
void out_gemm_kernel(const float* __restrict__ hidden,
                     const float* __restrict__ oW,
                     const float* __restrict__ ob,
                     float* __restrict__ out)
{
    const int wave = threadIdx.x >> 5;
    const int lane = threadIdx.x & 31;
    const int nl   = lane & 15;
    const int hi   = lane >> 4;

    const int tid  = blockIdx.x * 8 + wave;      // 0 .. 31999
    const int rg   = tid / 500;                  // row group: rows rg*64 ..
    const int cg   = tid % 500;                  // col group: cols cg*64 ..
    const int row0 = rg * 64;
    const int col0 = cg * 64;

    // Pull this wave's 16 KB B panel (64 rows of output_W) toward the WGP
    // while we load A tiles (global_prefetch_b8).
    __builtin_prefetch(&oW[(size_t)(col0 + lane) * DIM], 0, 3);
    __builtin_prefetch(&oW[(size_t)(col0 + 32 + lane) * DIM], 0, 3);

    // A-tiles: 4 M-tiles x 16 K-steps, reused across all 4 N-subtiles.
    v2f A[4][16];
    #pragma unroll
    for (int mt = 0; mt < 4; ++mt) {
        const float* hrow = hidden + (size_t)(row0 + mt * 16 + nl) * DIM;
        #pragma unroll
        for (int kk = 0; kk < 16; ++kk)
            A[mt][kk] = *(const v2f*)&hrow[4 * kk + 2 * hi];
    }

    #pragma unroll
    for (int s = 0; s < 4; ++s) {
        const int ncol = col0 + s * 16 + nl;     // column for this lane
        v8f acc[4] = {v8f{}, v8f{}, v8f{}, v8f{}};
        #pragma unroll
        for (int kk = 0; kk < 16; ++kk) {
            // B[d][n] = output_W[n][d]; one load feeds 4 WMMAs
            const v2f b = *(const v2f*)&oW[(size_t)ncol * DIM + 4 * kk + 2 * hi];
            #pragma unroll
            for (int mt = 0; mt < 4; ++mt)
                acc[mt] = wmma_f32_4(A[mt][kk], b, acc[mt]);
        }
        const float bias = ob[ncol];
        #pragma unroll
        for (int mt = 0; mt < 4; ++mt) {
            const int r0 = row0 + mt * 16 + 8 * hi;
            #pragma unroll
            for (int v = 0; v < 8; ++v)
                out[(size_t)(r0 + v) * VOCAB + ncol] = acc[mt][v] + bias;
        }
    }
}

extern "C" void kernel_launch(void* const* d_in, const int* in_sizes, int n_in,
                              void* d_out, int out_size, void* d_ws, size_t ws_size,
                              hipStream_t stream) {
    const int*   ids   = (const int*)  d_in[0];
    const float* embed = (const float*)d_in[1];
    const float* Ws    = (const float*)d_in[2];
    const float* rW    = (const float*)d_in[3];
    const float* rb    = (const float*)d_in[4];
    const float* cW    = (const float*)d_in[5];
    const float* cb    = (const float*)d_in[6];
    const float* oW    = (const float*)d_in[7];
    const float* ob    = (const float*)d_in[8];
    float* out    = (float*)d_out;
    float* hidden = (float*)d_ws;     // B*N*D f32 = 1 MB scratch

    moe_scan_kernel<<<1, 256, 0, stream>>>(ids, embed, Ws, rW, rb, cW, cb, hidden);
    out_gemm_kernel<<<4000, 256, 0, stream>>>(hidden, oW, ob, out);
}